// VQuantizer_59734405153291
// MI455X (gfx1250) — compile-verified
//
#include <hip/hip_runtime.h>

typedef float v2f __attribute__((ext_vector_type(2)));
typedef float v8f __attribute__((ext_vector_type(8)));

#define NTOK   65536    // 64*32*32 tokens
#define CCH    256      // channels
#define CNUM   1024     // codebook size
#define TPB    64       // tokens per workgroup (per batch: 1024, divisible)
#define PSTR   1025     // padded LDS row stride (1025 % 64 == 1 -> conflict-free)

// d_out float-element offsets (concatenated outputs, reference return order)
#define OUT_ZBAR 0
#define OUT_SOFT 16777216
#define OUT_HARD 33554432
#define OUT_SYM  50331648
#define OUT_PHI  50397184

static __device__ __forceinline__ v8f wmma_f32x4(v2f a, v2f b, v8f c) {
  // V_WMMA_F32_16X16X4_F32 : D = A(16x4,f32) * B(4x16,f32) + C(16x16,f32)
  return __builtin_amdgcn_wmma_f32_16x16x4_f32(
      false, a, false, b, (short)0, c, false, false);
}

// ---- tiny kernel: per-center squared norms ||e_j||^2 (1 MB, runs once) ----
__global__ void vq_enorm(const float* __restrict__ centers, float* __restrict__ e2) {
  int j = blockIdx.x * 256 + threadIdx.x;          // 4 blocks * 256 thr
  const float* p = centers + (size_t)j * CCH;
  float s = 0.f;
  for (int c = 0; c < CCH; ++c) { float v = p[c]; s += v * v; }
  e2[j] = s;
}

// ---- fused kernel: x-norms -> dist GEMM -> softmax/argmin -> phisoft ->
// ----               softout GEMM -> hardout/zbar gather ---------------------
__global__ void __launch_bounds__(256, 1)
vq_main(const float* __restrict__ data, const float* __restrict__ centers,
        const float* __restrict__ e2g, float* __restrict__ out) {
  __shared__ float P[TPB * PSTR];     // dist tile -> exp -> normalized phisoft
  __shared__ float x2s[TPB];
  __shared__ float e2s[CNUM];
  __shared__ float pmin[256];
  __shared__ int   pidx[256];
  __shared__ float psum[256];
  __shared__ float dminT[TPB];
  __shared__ int   symT[TPB];
  __shared__ float rden[TPB];

  const int tid  = threadIdx.x;
  const int wg   = blockIdx.x;
  const int n0   = wg * TPB;              // all 64 tokens share one batch
  const int b    = n0 >> 10;
  const int t0   = n0 & 1023;
  const float* Xb = data + (size_t)b * 262144 + t0;   // X(m,c) = Xb[c*1024+m]

  // ---- fused ||x||^2: 4 partial threads per token (X tile is hot anyway) --
  {
    const int m = tid & 63, part = tid >> 6;
    float s = 0.f;
    for (int c = part * 64; c < part * 64 + 64; ++c) {
      float v = Xb[(size_t)c * 1024 + m];
      s += v * v;
    }
    psum[part * 64 + m] = s;
  }
  for (int i = tid; i < CNUM; i += 256) e2s[i] = e2g[i];
  __syncthreads();
  if (tid < TPB)
    x2s[tid] = psum[tid] + psum[64 + tid] + psum[128 + tid] + psum[192 + tid];
  __syncthreads();

  const int wave = tid >> 5;
  const int lane = tid & 31;
  const int lo   = lane & 15;             // A: M offset / B: N offset
  const int hi   = lane >> 4;             // K half select
  const int m0   = (wave & 3) * 16;       // wave's M-tile
  const int half = wave >> 2;             // 2 waves share an M-tile

  float x2f[8];
  #pragma unroll
  for (int r = 0; r < 8; ++r) x2f[r] = x2s[m0 + r + 8 * hi];

  // ====== GEMM1: dist(m, j); 8 N-tiles in flight to amortize A loads ======
  for (int g = 0; g < 4; ++g) {
    const int jt = (half * 32 + g * 8) * 16;   // 8 consecutive 16-wide N tiles
    v8f acc[8] = {};
    for (int kk = 0; kk < CCH; kk += 4) {
      const int k = kk + 2 * hi;
      v2f a;
      a.x = Xb[(size_t)k * 1024 + m0 + lo];
      a.y = Xb[(size_t)(k + 1) * 1024 + m0 + lo];
      #pragma unroll
      for (int i = 0; i < 8; ++i) {
        v2f bi = *(const v2f*)(centers + (size_t)(jt + i * 16 + lo) * CCH + k);
        acc[i] = wmma_f32x4(a, bi, acc[i]);
      }
    }
    #pragma unroll
    for (int i = 0; i < 8; ++i) {
      const int j = jt + i * 16 + lo;
      const float e2v = e2s[j];
      #pragma unroll
      for (int r = 0; r < 8; ++r) {
        const int m = m0 + r + 8 * hi;
        float sq = x2f[r] - 2.0f * acc[i][r] + e2v;
        P[m * PSTR + j] = sqrtf(fmaxf(sq, 0.0f));     // dist
      }
    }
  }
  __syncthreads();

  // ================= softmax(-dist) + argmin(dist) ========================
  {  // pass A: partial min + first-index argmin (4 threads per token)
    const int m = tid & 63, part = tid >> 6, js = part * 256;
    float best = 3.4e38f; int bj = js;
    for (int jj = 0; jj < 256; ++jj) {
      float d = P[m * PSTR + js + jj];
      if (d < best) { best = d; bj = js + jj; }
    }
    pmin[part * 64 + m] = best; pidx[part * 64 + m] = bj;
  }
  __syncthreads();
  if (tid < TPB) {
    float best = pmin[tid]; int bj = pidx[tid];
    #pragma unroll
    for (int p = 1; p < 4; ++p) {
      float d = pmin[p * 64 + tid];
      if (d < best) { best = d; bj = pidx[p * 64 + tid]; }
    }
    dminT[tid] = best; symT[tid] = bj;
  }
  __syncthreads();
  {  // pass B: exp(dmin - d) cached in LDS + partial sums
    const int m = tid & 63, part = tid >> 6, js = part * 256;
    const float dm = dminT[m];
    float s = 0.f;
    for (int jj = 0; jj < 256; ++jj) {
      const int idx = m * PSTR + js + jj;
      float e = __expf(dm - P[idx]);
      P[idx] = e; s += e;
    }
    psum[part * 64 + m] = s;
  }
  __syncthreads();
  if (tid < TPB) {
    float s = psum[tid] + psum[64 + tid] + psum[128 + tid] + psum[192 + tid];
    rden[tid] = 1.0f / s;
    ((int*)(out + OUT_SYM))[n0 + tid] = symT[tid];   // raw int32 symbols
  }
  __syncthreads();
  {  // pass C: normalize in LDS and stream phisoft to HBM (token-coalesced)
    float* phi = out + OUT_PHI + (size_t)b * 1048576 + t0;
    const int m = tid & 63;
    const float rd = rden[m];
    for (int j0 = 0; j0 < CNUM; j0 += 4) {
      const int jj = j0 + (tid >> 6);
      float v = P[m * PSTR + jj] * rd;
      P[m * PSTR + jj] = v;
      phi[(size_t)jj * 1024 + m] = v;
    }
  }
  __syncthreads();

  // ====== GEMM2: softout = phisoft @ centers; 8 c-tiles share A frag ======
  {
    float* so = out + OUT_SOFT + (size_t)b * 262144 + t0;
    const int cbase = half * 128;                 // 8 tiles of 16 channels
    v8f acc[8] = {};
    for (int kk = 0; kk < CNUM; kk += 4) {
      const int k = kk + 2 * hi;
      v2f a;
      a.x = P[(m0 + lo) * PSTR + k];
      a.y = P[(m0 + lo) * PSTR + k + 1];
      #pragma unroll
      for (int q = 0; q < 8; ++q) {
        const int c = cbase + q * 16 + lo;
        v2f bb;
        bb.x = centers[(size_t)k * CCH + c];
        bb.y = centers[(size_t)(k + 1) * CCH + c];
        acc[q] = wmma_f32x4(a, bb, acc[q]);
      }
    }
    #pragma unroll
    for (int q = 0; q < 8; ++q) {
      const int c = cbase + q * 16 + lo;
      #pragma unroll
      for (int r = 0; r < 8; ++r)
        so[(size_t)c * 1024 + m0 + r + 8 * hi] = acc[q][r];
    }
  }

  // ================= hardout gather; zbar == hardout =====================
  {
    float* hb = out + OUT_HARD + (size_t)b * 262144 + t0;
    float* zb = out + OUT_ZBAR + (size_t)b * 262144 + t0;
    const int m = tid & 63;
    const int sym = symT[m];
    for (int c0 = 0; c0 < CCH; c0 += 4) {
      const int c = c0 + (tid >> 6);
      float v = centers[(size_t)sym * CCH + c];
      hb[(size_t)c * 1024 + m] = v;
      zb[(size_t)c * 1024 + m] = v;
    }
  }
}

extern "C" void kernel_launch(void* const* d_in, const int* in_sizes, int n_in,
                              void* d_out, int out_size, void* d_ws, size_t ws_size,
                              hipStream_t stream) {
  (void)in_sizes; (void)n_in; (void)out_size; (void)ws_size;
  const float* data    = (const float*)d_in[0];
  const float* centers = (const float*)d_in[1];
  float* e2  = (float*)d_ws;         // 1024 floats of scratch
  float* out = (float*)d_out;

  vq_enorm<<<CNUM / 256, 256, 0, stream>>>(centers, e2);
  vq_main<<<NTOK / TPB, 256, 0, stream>>>(data, centers, e2, out);
}